// TargetProbability_77816217469032
// MI455X (gfx1250) — compile-verified
//
#include <hip/hip_runtime.h>

typedef float v2f __attribute__((ext_vector_type(2)));
typedef float v8f __attribute__((ext_vector_type(8)));
typedef unsigned int u32x4 __attribute__((ext_vector_type(4)));
typedef int i32x4 __attribute__((ext_vector_type(4)));
typedef int i32x8 __attribute__((ext_vector_type(8)));

namespace {
constexpr int kB = 4, kS = 128, kV = 50257, kD = 1024;
constexpr int kM = kB * kS;                 // 512 token rows
constexpr int MT = 64;                      // M tile per workgroup
constexpr int NT = 128;                     // N tile per workgroup
constexpr int KC = 32;                      // K chunk staged via TDM into LDS
constexpr int KSPLIT = 8;                   // split-K for occupancy (512 WGs)
constexpr int KLEN = 6304;                  // per-split K length (mult of KC; 8*6304 >= kV)
constexpr int KPAD = ((kV + KC - 1) / KC) * KC;  // 50272
// LDS strides produced by TDM padding: pad every row to break bank conflicts.
constexpr int A_STRIDE = KC + 2;            // 34 dwords  (pad_interval 32 dw, pad 2 dw)
constexpr int B_STRIDE = NT + 4;            // 132 dwords (pad_interval 128 dw, pad 4 dw)
constexpr int A_DW = MT * A_STRIDE;         // 2176 dwords per buffer
constexpr int B_DW = KC * B_STRIDE;         // 4224 dwords per buffer
constexpr size_t EMB_OFF = 0;
constexpr size_t PRED_OFF = (size_t)kM * kD;          // 524288
constexpr size_t PROBS_OFF = PRED_OFF + (size_t)kM;   // 524800
}

// Issue one 2-D TDM tile load: tile [tile_d1 rows x tile_d0 elems] of f32 from
// global (row stride stride0 elems) into LDS at lds_addr, inserting pad_amount
// dwords of padding after every 2^(pad_iv+1) dwords (i.e., per row).
// Rows/elems beyond tensor_d1/tensor_d0 are zero-filled by the TDM.
__device__ __forceinline__ void tdm_load_2d(
    unsigned lds_addr, const void* gptr,
    unsigned tensor_d0, unsigned tensor_d1, unsigned long long stride0,
    unsigned tile_d0, unsigned tile_d1,
    unsigned pad_iv, unsigned pad_amt)   // encoded fields
{
  unsigned long long ga = (unsigned long long)(size_t)gptr;
  u32x4 g0 = {
    1u,                                             // count=1 (valid, user mode)
    lds_addr,                                       // bits 63:32 lds_addr
    (unsigned)ga,                                   // global_addr lo
    ((unsigned)(ga >> 32) & 0x01FFFFFFu) | (2u << 30) // global_addr hi | type=2
  };
  i32x8 g1 = {
    (int)((2u << 16) | (1u << 20) | (pad_iv << 22) | (pad_amt << 25)),
    (int)((tensor_d0 & 0xFFFFu) << 16),             // [63:48] tensor_dim0 lo16
    (int)((tensor_d0 >> 16) | ((tensor_d1 & 0xFFFFu) << 16)),
    (int)((tensor_d1 >> 16) | (tile_d0 << 16)),     // tensor_dim1 hi | tile_dim0
    (int)(tile_d1 & 0xFFFFu),                       // tile_dim1; tile_dim2=0
    (int)(unsigned)(stride0 & 0xFFFFFFFFull),       // dim0_stride lo32
    (int)(unsigned)((stride0 >> 32) & 0xFFFFull),   // dim0_stride hi16; dim1_stride lo=0
    0
  };
  i32x4 gz4 = {0, 0, 0, 0};
  i32x8 gz8 = {0, 0, 0, 0, 0, 0, 0, 0};
  __builtin_amdgcn_tensor_load_to_lds(g0, g1, gz4, gz4, gz8, 0);
}

// D[M,N] (+)= probs[M,K] * embed[K,N] over K range [z*klen, z*klen+klen).
// TDM-fed, double-buffered LDS, f32 WMMA 16x16x4.
__global__ __launch_bounds__(256) void gemm_wmma_f32(
    const float* __restrict__ probs, const float* __restrict__ embed,
    float* __restrict__ out, int klen)
{
  __shared__ __align__(16) float smem[2 * A_DW + 2 * B_DW];   // ~51 KB

  const int tid  = threadIdx.x;
  const int lane = tid & 31;
  const int wave = tid >> 5;         // 0..7
  const int wm   = (wave >> 2) * 32; // wave M offset: 0/32
  const int wn   = (wave & 3) * 32;  // wave N offset: 0/32/64/96
  const int lh   = lane >> 4;        // half-wave (K pair select)
  const int ll   = lane & 15;

  const int m0 = blockIdx.y * MT;
  const int n0 = blockIdx.x * NT;
  const int kbeg = blockIdx.z * klen;
  int kend = kbeg + klen;
  if (kend > KPAD) kend = KPAD;
  const int nch = (kend - kbeg) / KC;

  // Issue TDM loads for chunk c into buffer buf (wave-uniform; wave 0 only).
  // LDS byte address = low 32 bits of the generic address (runtime scalar cast;
  // no constant aggregates of addrspacecast expressions).
  auto issue = [&](int c, int buf) {
    const unsigned ldsBase = (unsigned)(size_t)(void*)&smem[0];
    const unsigned aAddr = ldsBase + (unsigned)(buf * A_DW) * 4u;
    const unsigned bAddr = ldsBase + (unsigned)(2 * A_DW + buf * B_DW) * 4u;
    const int k0 = kbeg + c * KC;
    // A tile: rows = m (MT of them, stride kV), elems = k (KC), zero-fill k>=kV.
    tdm_load_2d(aAddr, probs + (size_t)m0 * kV + k0,
                (unsigned)(kV - k0), (unsigned)(kM - m0), (unsigned long long)kV,
                KC, MT, /*pad 2^(4+1)=32dw*/4u, /*2dw*/1u);
    // B tile: rows = k (KC of them, stride kD), elems = n (NT), zero-fill k>=kV.
    tdm_load_2d(bAddr, embed + (size_t)k0 * kD + n0,
                (unsigned)(kD - n0), (unsigned)(kV - k0), (unsigned long long)kD,
                NT, KC, /*pad 2^(6+1)=128dw*/6u, /*4dw*/3u);
  };

  v8f acc[2][2] = {};

  if (wave == 0) {
    issue(0, 0);
    if (nch > 1) issue(1, 1);
  }

  for (int c = 0; c < nch; ++c) {
    const int cur = c & 1;
    if (wave == 0) {
      if (c + 1 < nch) __builtin_amdgcn_s_wait_tensorcnt(2);  // chunk c done, c+1 in flight
      else             __builtin_amdgcn_s_wait_tensorcnt(0);
    }
    __syncthreads();                      // chunk c visible to all waves

    const float* As = smem + cur * A_DW;
    const float* Bs = smem + 2 * A_DW + cur * B_DW;
    #pragma unroll
    for (int kk = 0; kk < KC; kk += 4) {
      v2f a[2], b[2];
      #pragma unroll
      for (int mi = 0; mi < 2; ++mi)
        // A 16x4 frag: lane<16 holds (K=kk,kk+1), lane>=16 holds (kk+2,kk+3);
        // row-major padded LDS makes the pair one aligned b64 load.
        a[mi] = *(const v2f*)(As + (size_t)(wm + mi * 16 + ll) * A_STRIDE + kk + lh * 2);
      #pragma unroll
      for (int ni = 0; ni < 2; ++ni) {
        b[ni].x = Bs[(size_t)(kk + lh * 2 + 0) * B_STRIDE + wn + ni * 16 + ll];
        b[ni].y = Bs[(size_t)(kk + lh * 2 + 1) * B_STRIDE + wn + ni * 16 + ll];
      }
      #pragma unroll
      for (int mi = 0; mi < 2; ++mi)
        #pragma unroll
        for (int ni = 0; ni < 2; ++ni)
          acc[mi][ni] = __builtin_amdgcn_wmma_f32_16x16x4_f32(
              false, a[mi], false, b[ni], (short)0, acc[mi][ni], false, false);
    }
    __syncthreads();                      // everyone done reading buffer cur
    if (wave == 0 && c + 2 < nch) issue(c + 2, cur);   // refill freed buffer
  }

  // Epilogue: C/D layout — VGPR r: lanes 0-15 -> M=r, lanes 16-31 -> M=8+r.
  float* dst = out + (size_t)blockIdx.z * kM * kD;
  #pragma unroll
  for (int mi = 0; mi < 2; ++mi)
    #pragma unroll
    for (int ni = 0; ni < 2; ++ni)
      #pragma unroll
      for (int r = 0; r < 8; ++r) {
        const int row = m0 + wm + mi * 16 + lh * 8 + r;
        const int col = n0 + wn + ni * 16 + ll;
        dst[(size_t)row * kD + col] = acc[mi][ni][r];
      }
}

// Deterministic fixed-order sum of KSPLIT partials.
__global__ __launch_bounds__(256) void reduce_split(
    const float* __restrict__ part, float* __restrict__ out)
{
  const int i = blockIdx.x * 256 + threadIdx.x;
  if (i < kM * kD) {
    float s = 0.f;
    #pragma unroll
    for (int z = 0; z < KSPLIT; ++z) s += part[(size_t)z * kM * kD + i];
    out[EMB_OFF + i] = s;
  }
}

// Fused: copy pred_probs to output + argmax per (b,s) row (first-max tie-break).
__global__ __launch_bounds__(256) void argmax_copy(
    const float* __restrict__ probs, float* __restrict__ out)
{
  __shared__ float sv[256];
  __shared__ int   si[256];
  const int row = blockIdx.x;
  const int tid = threadIdx.x;
  const float* src = probs + (size_t)row * kV;
  float* dstp = out + PROBS_OFF + (size_t)row * kV;

  float best = -3.402823466e38f;
  int bidx = 0;
  for (int i = tid; i < kV; i += 256) {
    const float v = src[i];
    dstp[i] = v;
    if (v > best) { best = v; bidx = i; }   // ascending scan keeps first max
  }
  sv[tid] = best; si[tid] = bidx;
  __syncthreads();
  for (int off = 128; off > 0; off >>= 1) {
    if (tid < off) {
      const float ov = sv[tid + off]; const int oi = si[tid + off];
      if (ov > sv[tid] || (ov == sv[tid] && oi < si[tid])) { sv[tid] = ov; si[tid] = oi; }
    }
    __syncthreads();
  }
  if (tid == 0) out[PRED_OFF + row] = (float)si[0];
}

extern "C" void kernel_launch(void* const* d_in, const int* in_sizes, int n_in,
                              void* d_out, int out_size, void* d_ws, size_t ws_size,
                              hipStream_t stream) {
  (void)in_sizes; (void)n_in; (void)out_size;
  const float* probs = (const float*)d_in[0];   // [B,S,V] f32
  const float* embed = (const float*)d_in[1];   // [V,D] f32
  float* out = (float*)d_out;                   // [emb | pred | probs] f32

  dim3 blk(256);
  const size_t need = (size_t)KSPLIT * kM * kD * sizeof(float);
  if (ws_size >= need) {
    dim3 grid(kD / NT, kM / MT, KSPLIT);        // 8 x 8 x 8 = 512 workgroups
    gemm_wmma_f32<<<grid, blk, 0, stream>>>(probs, embed, (float*)d_ws, KLEN);
    reduce_split<<<(kM * kD + 255) / 256, blk, 0, stream>>>((const float*)d_ws, out);
  } else {
    dim3 grid(kD / NT, kM / MT, 1);
    gemm_wmma_f32<<<grid, blk, 0, stream>>>(probs, embed, out, KPAD);
  }
  argmax_copy<<<kM, blk, 0, stream>>>(probs, out);
}